// TransitionModel_24945170055310
// MI455X (gfx1250) — compile-verified
//
#include <hip/hip_runtime.h>
#include <math.h>

// ---------------------------------------------------------------------------
// Particle-filter transition model (sampling motion model), MI455X / gfx1250.
//
// Roofline: 230.4 MB streamed once / 23.3 TB/s ~= 9.9 us -> HBM bound.
// Strategy: pure streaming kernel, 128-bit coalesced non-temporal loads/stores
// (footprint > 192MB L2, zero reuse), native v_sin/v_cos, closed-form angle
// wrap (d - 2*pi*rint(d/2pi)) instead of per-element software atan2.
// All odometry-only scalar math is hoisted into a 1-thread prologue kernel
// writing 6 floats into d_ws.
// ---------------------------------------------------------------------------

typedef __attribute__((ext_vector_type(4))) float v4f;

#define TM_TWO_PI      6.28318530717958647692f
#define TM_INV_TWO_PI  0.15915494309189533577f
#define TM_ALPHA       0.1f

// Exact closed form of atan2(sin(d), cos(d)): wrap d into [-pi, pi].
// v_mul + v_rndne_f32 + v_fma -- no software atan2 expansion.
__device__ __forceinline__ float tm_wrap_pi(float d) {
    return __builtin_fmaf(-TM_TWO_PI, rintf(d * TM_INV_TWO_PI), d);
}

// --- Prologue: fold odometry/old_pose into 6 uniform scalars in d_ws. ------
// params = { delta_rot1, delta_trans, delta_rot2,
//            rot1_scale, trans_scale, rot2_scale }
__global__ void tm_setup_params(const float* __restrict__ odom,
                                const float* __restrict__ old_pose,
                                float* __restrict__ params) {
    if (blockIdx.x != 0 || threadIdx.x != 0) return;
    float ax  = odom[0];
    float ay  = odom[1];
    float ath = odom[2];
    float th1 = old_pose[2];

    float dt  = sqrtf(ax * ax + ay * ay);
    float dr1 = (dt < 0.01f) ? 0.0f : tm_wrap_pi(atan2f(ay, ax) - th1);
    float dr2 = tm_wrap_pi(ath - dr1);

    params[0] = dr1;
    params[1] = dt;
    params[2] = dr2;
    params[3] = TM_ALPHA * (dr1 * dr1 + dt * dt);              // rot1_scale
    params[4] = TM_ALPHA * (dt * dt + dr1 * dr1 + dr2 * dr2);  // trans_scale
    params[5] = TM_ALPHA * (dr2 * dr2 + dt * dt);              // rot2_scale
}

// --- Main streaming kernel: 4 particles / thread, all-b128 NT traffic. -----
__global__ __launch_bounds__(256) void tm_transition(
    const float* __restrict__ ps,      // [n,3] particle states (x,y,th)
    const float* __restrict__ er1,     // [n] eps_rot1
    const float* __restrict__ etr,     // [n] eps_trans
    const float* __restrict__ er2,     // [n] eps_rot2
    const float* __restrict__ params,  // [6] uniform scalars
    float*       __restrict__ out,     // [n,3]
    int n)                             // particle count (B*P)
{
    const int t    = blockIdx.x * blockDim.x + threadIdx.x;
    const int base = t * 4;
    if (base >= n) return;

    // Uniform scalars (address is wave-uniform; cheap cached loads).
    const float dr1 = params[0];
    const float dt  = params[1];
    const float dr2 = params[2];
    const float s1  = params[3];
    const float st  = params[4];
    const float s2  = params[5];

    if (base + 4 <= n) {
        // Fast path: 6x global_load_b128 NT + 3x global_store_b128 NT,
        // all 16B-aligned and wave-contiguous.
        const v4f* ps4 = (const v4f*)ps;
        const v4f  a   = __builtin_nontemporal_load(ps4 + 3 * t + 0);
        const v4f  b   = __builtin_nontemporal_load(ps4 + 3 * t + 1);
        const v4f  c   = __builtin_nontemporal_load(ps4 + 3 * t + 2);
        const v4f  e1  = __builtin_nontemporal_load((const v4f*)er1 + t);
        const v4f  et  = __builtin_nontemporal_load((const v4f*)etr + t);
        const v4f  e2  = __builtin_nontemporal_load((const v4f*)er2 + t);

        // Deinterleave AoS (x,y,th) in registers.
        const float X[4] = {a[0], a[3], b[2], c[1]};
        const float Y[4] = {a[1], b[0], b[3], c[2]};
        const float T[4] = {a[2], b[1], c[0], c[3]};
        float OX[4], OY[4], OT[4];

#pragma unroll
        for (int j = 0; j < 4; ++j) {
            const float r1h = tm_wrap_pi(dr1 - e1[j] * s1);
            const float trh = __builtin_fmaf(-et[j], st, dt);
            const float r2h = tm_wrap_pi(dr2 - e2[j] * s2);
            float sn, cs;
            __sincosf(T[j] + r1h, &sn, &cs);        // native v_sin/v_cos
            OX[j] = __builtin_fmaf(trh, cs, X[j]);
            OY[j] = __builtin_fmaf(trh, sn, Y[j]);
            OT[j] = T[j] + r1h + r2h;
        }

        const v4f o0 = {OX[0], OY[0], OT[0], OX[1]};
        const v4f o1 = {OY[1], OT[1], OX[2], OY[2]};
        const v4f o2 = {OT[2], OX[3], OY[3], OT[3]};
        v4f* out4 = (v4f*)out;
        __builtin_nontemporal_store(o0, out4 + 3 * t + 0);
        __builtin_nontemporal_store(o1, out4 + 3 * t + 1);
        __builtin_nontemporal_store(o2, out4 + 3 * t + 2);
    } else {
        // Tail (only if n % 4 != 0; n = 6.4M is divisible, but stay generic).
        for (int p = base; p < n; ++p) {
            const float x  = ps[3 * p + 0];
            const float y  = ps[3 * p + 1];
            const float th = ps[3 * p + 2];
            const float r1h = tm_wrap_pi(dr1 - er1[p] * s1);
            const float trh = __builtin_fmaf(-etr[p], st, dt);
            const float r2h = tm_wrap_pi(dr2 - er2[p] * s2);
            float sn, cs;
            __sincosf(th + r1h, &sn, &cs);
            out[3 * p + 0] = __builtin_fmaf(trh, cs, x);
            out[3 * p + 1] = __builtin_fmaf(trh, sn, y);
            out[3 * p + 2] = th + r1h + r2h;
        }
    }
}

extern "C" void kernel_launch(void* const* d_in, const int* in_sizes, int n_in,
                              void* d_out, int out_size, void* d_ws, size_t ws_size,
                              hipStream_t stream) {
    // setup_inputs() order:
    //   0: particle_states [B,P,3] f32
    //   1: odometry        [3]     f32
    //   2: old_pose        [3]     f32
    //   3: eps_rot1        [B,P]   f32
    //   4: eps_trans       [B,P]   f32
    //   5: eps_rot2        [B,P]   f32
    const float* ps       = (const float*)d_in[0];
    const float* odom     = (const float*)d_in[1];
    const float* old_pose = (const float*)d_in[2];
    const float* e1       = (const float*)d_in[3];
    const float* et       = (const float*)d_in[4];
    const float* e2       = (const float*)d_in[5];
    float*       out      = (float*)d_out;
    float*       params   = (float*)d_ws;   // 6 floats of scratch

    const int n  = in_sizes[0] / 3;         // B*P particles
    const int n4 = (n + 3) / 4;             // 4 particles per thread
    const int threads = 256;                // 8 wave32s per block
    const int blocks  = (n4 + threads - 1) / threads;

    tm_setup_params<<<1, 32, 0, stream>>>(odom, old_pose, params);
    tm_transition<<<blocks, threads, 0, stream>>>(ps, e1, et, e2, params, out, n);
}